// SimpleTTS_42468636623484
// MI455X (gfx1250) — compile-verified
//
#include <hip/hip_runtime.h>
#include <hip/hip_bf16.h>
#include <cstddef>

// ---------------------------------------------------------------------------
// SimpleTTS decoder for MI455X (gfx1250, wave32).
//  * Encoder is dead code (output unused) -> skipped.
//  * Recurrence folded:  g_t = h_{t-1} @ (Whh + Wih@lin_W)^T + (b + Wih@lin_b)
//  * fp32 WMMA (V_WMMA_F32_16X16X4_F32): latency-bound problem, precision free.
//  * 64 persistent workgroups; each stages its 256KB W_fold slice into the
//    320KB WGP LDS once; grid barrier per step via agent-scope atomics.
// ---------------------------------------------------------------------------

typedef float v2f __attribute__((ext_vector_type(2)));
typedef float v8f __attribute__((ext_vector_type(8)));

#define BATCH   64
#define HID     1024
#define G4      4096
#define MEL     80
#define NBLK    64
#define NTHR    288            // 9 waves: 8 gate waves + 1 frame wave

// workspace layout (floats)
#define WS_BSWZ 0                           // 4096*1024 swizzled W_fold
#define WS_LSWZ (WS_BSWZ + G4*HID)          // 5*128*32*4 swizzled lin_W^T
#define WS_B2   (WS_LSWZ + 81920)           // 4096 folded bias
#define WS_HBUF (WS_B2 + G4)                // 2 * 64*1024 double-buffered h
#define WS_BAR  (WS_HBUF + 2*BATCH*HID)     // 2 u32 barrier state

__device__ __forceinline__ v8f wmma4(v2f a, v2f b, v8f c) {
  // D = A(16x4 f32) * B(4x16 f32) + C(16x16 f32)
  return __builtin_amdgcn_wmma_f32_16x16x4_f32(false, a, false, b,
                                               (short)0, c, false, false);
}

__device__ __forceinline__ float sigm(float x) { return 1.f / (1.f + expf(-x)); }

__device__ __forceinline__ void grid_barrier(unsigned* bar) {
  __syncthreads();
  if (threadIdx.x == 0) {
    __threadfence();
    unsigned* cnt = bar;
    unsigned* gen = bar + 1;
    unsigned g = __hip_atomic_load(gen, __ATOMIC_RELAXED, __HIP_MEMORY_SCOPE_AGENT);
    unsigned a = __hip_atomic_fetch_add(cnt, 1u, __ATOMIC_ACQ_REL, __HIP_MEMORY_SCOPE_AGENT);
    if (a == NBLK - 1) {
      __hip_atomic_store(cnt, 0u, __ATOMIC_RELAXED, __HIP_MEMORY_SCOPE_AGENT);
      __hip_atomic_store(gen, g + 1u, __ATOMIC_RELEASE, __HIP_MEMORY_SCOPE_AGENT);
    } else {
      while (__hip_atomic_load(gen, __ATOMIC_ACQUIRE, __HIP_MEMORY_SCOPE_AGENT) == g)
        __builtin_amdgcn_s_sleep(1);
    }
  }
  __syncthreads();
}

// --------------------------- prep: fold + swizzle ---------------------------
// Bswz element e = ((nt*128+kp)*32+lane)*4+c holds
//   W_fold[N][K],  N = 16*nt + (lane&15),  K = 4*(2*kp + (c>>1)) + 2*(lane>>4) + (c&1)
__global__ void tts_prep(const float* __restrict__ dWih, const float* __restrict__ dWhh,
                         const float* __restrict__ dbih, const float* __restrict__ dbhh,
                         const float* __restrict__ linW, const float* __restrict__ linb,
                         float* __restrict__ ws) {
  float* Bswz = ws + WS_BSWZ;
  float* Lswz = ws + WS_LSWZ;
  float* b2   = ws + WS_B2;
  unsigned* bar = (unsigned*)(ws + WS_BAR);
  long gid = (long)blockIdx.x * blockDim.x + threadIdx.x;
  long gsz = (long)gridDim.x * blockDim.x;
  if (gid == 0) { bar[0] = 0u; bar[1] = 0u; }

  for (long e = gid; e < (long)G4 * HID; e += gsz) {
    int c = e & 3, lane = (e >> 2) & 31, kp = (e >> 7) & 127, nt = (int)(e >> 14);
    int K = 4 * (2 * kp + (c >> 1)) + 2 * (lane >> 4) + (c & 1);
    int N = 16 * nt + (lane & 15);
    float acc = dWhh[(long)N * HID + K];
    for (int m = 0; m < MEL; ++m)
      acc = fmaf(dWih[N * MEL + m], linW[m * HID + K], acc);
    Bswz[e] = acc;
  }
  for (long e = gid; e < 81920; e += gsz) {
    int c = e & 3, lane = (e >> 2) & 31, kp = (e >> 7) & 127, nt = (int)(e >> 14);
    int K = 4 * (2 * kp + (c >> 1)) + 2 * (lane >> 4) + (c & 1);
    int N = 16 * nt + (lane & 15);
    Lswz[e] = (N < MEL) ? linW[N * HID + K] : 0.f;
  }
  for (long e = gid; e < G4; e += gsz) {
    float acc = dbih[e] + dbhh[e];
    for (int m = 0; m < MEL; ++m)
      acc = fmaf(dWih[e * MEL + m], linb[m], acc);
    b2[e] = acc;
  }
}

// ------------------------------ frame output -------------------------------
__device__ __forceinline__ void frame_out(const float* __restrict__ hc,
                                          const float* __restrict__ Lswz,
                                          const float* __restrict__ linb,
                                          float* __restrict__ out,
                                          int wg, int lane, int t, int T) {
  int mt = wg & 3, nt = wg >> 2;                       // 20 tiles cover 64x80
  const float* hrow = hc + (mt * 16 + (lane & 15)) * HID + 2 * (lane >> 4);
  const float4* Lf  = (const float4*)Lswz + nt * 4096 + lane;
  v8f acc = {};
  for (int kp = 0; kp < 128; ++kp) {
    v2f a0 = *(const v2f*)(hrow + 8 * kp);
    v2f a1 = *(const v2f*)(hrow + 8 * kp + 4);
    float4 w = Lf[(size_t)kp * 32];
    v2f b0 = {w.x, w.y}, b1 = {w.z, w.w};
    acc = wmma4(a0, b0, acc);
    acc = wmma4(a1, b1, acc);
  }
  int half = lane >> 4, n = lane & 15, col = nt * 16 + n;
  float lb = linb[col];
  for (int v = 0; v < 8; ++v) {
    int b = mt * 16 + v + 8 * half;
    out[(size_t)b * T * MEL + (size_t)t * MEL + col] = acc[v] + lb;
  }
}

// --------------------------- persistent decoder ----------------------------
__global__ void __launch_bounds__(NTHR, 1)
tts_decode(const float* __restrict__ dbih, const float* __restrict__ dbhh,
           const int* __restrict__ Tptr, const float* __restrict__ linb,
           float* __restrict__ ws, float* __restrict__ out) {
  extern __shared__ float lds[];
  float* Bs = lds;                 // 4 tiles * 128kp * 32 lanes * 4 = 65536 f
  float* Gs = lds + 65536;         // 64 batch x 64 gate-cols staging
  float* Bswz = ws + WS_BSWZ;
  float* Lswz = ws + WS_LSWZ;
  float* b2   = ws + WS_B2;
  float* hbuf = ws + WS_HBUF;
  unsigned* bar = (unsigned*)(ws + WS_BAR);

  const int T    = Tptr[0];
  const int wg   = blockIdx.x;          // owns hidden slice [16*wg, 16*wg+16)
  const int tid  = threadIdx.x;
  const int wave = tid >> 5, lane = tid & 31;
  const int j0   = wg * 16;

  // Stage this WG's 4 gate B-tiles (gate g -> global n-tile 64*g + wg) to LDS.
  for (int i = tid; i < 16384; i += NTHR) {
    int g = i >> 12, rem = i & 4095;
    ((float4*)Bs)[i] = ((const float4*)Bswz)[(g * 64 + wg) * 4096 + rem];
  }
  __syncthreads();

  // t = 0: h=c=x=0  =>  g = bih + bhh  (elementwise only)
  float c_reg[4] = {0.f, 0.f, 0.f, 0.f};
  if (tid < 256) {
    for (int r = 0; r < 4; ++r) {
      int cell = tid + 256 * r;
      int b = cell >> 4, jl = cell & 15, j = j0 + jl;
      float gi = dbih[j]            + dbhh[j];
      float gg = dbih[2 * HID + j]  + dbhh[2 * HID + j];
      float go = dbih[3 * HID + j]  + dbhh[3 * HID + j];
      float c  = sigm(gi) * tanhf(gg);            // f-gate * c(=0) dropped
      c_reg[r] = c;
      hbuf[b * HID + j] = sigm(go) * tanhf(c);    // h_0 -> buffer 0
    }
  }
  grid_barrier(bar);

  for (int t = 1; t < T; ++t) {
    const float* hc = hbuf + ((t - 1) & 1) * (BATCH * HID);
    float*       hn = hbuf + (t & 1) * (BATCH * HID);

    if (wave < 8) {
      // gate GEMM: 2 C-tiles per wave, K=1024 via 128 x (2 WMMA_F32_16X16X4)
      int mt = wave & 3, p = wave >> 2;             // gates {2p, 2p+1}
      const float* hrow = hc + (mt * 16 + (lane & 15)) * HID + 2 * (lane >> 4);
      const float4* B0 = (const float4*)Bs + (2 * p) * 4096 + lane;
      const float4* B1 = (const float4*)Bs + (2 * p + 1) * 4096 + lane;
      v8f acc0 = {}, acc1 = {};
      for (int kp = 0; kp < 128; ++kp) {
        v2f a0 = *(const v2f*)(hrow + 8 * kp);
        v2f a1 = *(const v2f*)(hrow + 8 * kp + 4);
        float4 w0 = B0[(size_t)kp * 32];
        float4 w1 = B1[(size_t)kp * 32];
        v2f b00 = {w0.x, w0.y}, b01 = {w0.z, w0.w};
        v2f b10 = {w1.x, w1.y}, b11 = {w1.z, w1.w};
        acc0 = wmma4(a0, b00, acc0);
        acc0 = wmma4(a1, b01, acc0);
        acc1 = wmma4(a0, b10, acc1);
        acc1 = wmma4(a1, b11, acc1);
      }
      int half = lane >> 4, n = lane & 15;
      for (int v = 0; v < 8; ++v) {
        int b = mt * 16 + v + 8 * half;
        Gs[b * 64 + (2 * p) * 16 + n]     = acc0[v];
        Gs[b * 64 + (2 * p + 1) * 16 + n] = acc1[v];
      }
    } else if (wg < 20) {
      // frame wave: emit frame_{t-1} = h_{t-1} @ lin_W^T + lin_b, hidden
      // under the (2x longer) gate GEMM of waves 0..7.
      frame_out(hc, Lswz, linb, out, wg, lane, t - 1, T);
    }
    __syncthreads();

    // LSTM cell elementwise; c lives in registers, h published to global.
    if (tid < 256) {
      for (int r = 0; r < 4; ++r) {
        int cell = tid + 256 * r;
        int b = cell >> 4, jl = cell & 15, j = j0 + jl;
        float gi = Gs[b * 64 + jl]      + b2[j];
        float gf = Gs[b * 64 + 16 + jl] + b2[HID + j];
        float gg = Gs[b * 64 + 32 + jl] + b2[2 * HID + j];
        float go = Gs[b * 64 + 48 + jl] + b2[3 * HID + j];
        float c  = sigm(gf) * c_reg[r] + sigm(gi) * tanhf(gg);
        c_reg[r] = c;
        hn[b * HID + j] = sigm(go) * tanhf(c);
      }
    }
    grid_barrier(bar);   // publishes h_t; also fences frame reads of h_{t-1}
  }

  // last frame: frame_{T-1} from h_{T-1} (already fenced by final barrier)
  if (wave == 8 && wg < 20) {
    const float* hc = hbuf + ((T - 1) & 1) * (BATCH * HID);
    frame_out(hc, Lswz, linb, out, wg, lane, T - 1, T);
  }
}

// ------------------------------- launch ------------------------------------
extern "C" void kernel_launch(void* const* d_in, const int* in_sizes, int n_in,
                              void* d_out, int out_size, void* d_ws, size_t ws_size,
                              hipStream_t stream) {
  (void)in_sizes; (void)n_in; (void)out_size; (void)ws_size;
  const int*   Tptr = (const int*)d_in[2];
  const float* dWih = (const float*)d_in[8];
  const float* dWhh = (const float*)d_in[9];
  const float* dbih = (const float*)d_in[10];
  const float* dbhh = (const float*)d_in[11];
  const float* linW = (const float*)d_in[12];
  const float* linb = (const float*)d_in[13];
  float* out = (float*)d_out;
  float* ws  = (float*)d_ws;

  tts_prep<<<2048, 256, 0, stream>>>(dWih, dWhh, dbih, dbhh, linW, linb, ws);

  size_t lds_bytes = (size_t)(65536 + 64 * 64) * sizeof(float);  // 272 KB/WGP
  tts_decode<<<NBLK, NTHR, lds_bytes, stream>>>(dbih, dbhh, Tptr, linb, ws, out);
}